// GATClassifier_44152263803039
// MI455X (gfx1250) — compile-verified
//
#include <hip/hip_runtime.h>
#include <float.h>
#include <math.h>

#define NN 50000
#define EE 800000
#define GG 64
#define DD 128
#define HH 8
#define CC 16
#define NCLS 10
#define SLOPE 0.2f
#define BN_EPS 1e-5f

#define ETOT (EE + NN)          // edges incl. self loops
#define ND ((size_t)NN * DD)    // 6,400,000

typedef __attribute__((ext_vector_type(2))) float v2f;
typedef __attribute__((ext_vector_type(8))) float v8f;

// ---------------------------------------------------------------- utilities
__device__ __forceinline__ float leaky(float x) { return x > 0.f ? x : SLOPE * x; }
__device__ __forceinline__ float elu(float x)   { return x > 0.f ? x : expf(x) - 1.f; }

// race-free float atomic max via signed/unsigned integer ordering trick
__device__ __forceinline__ void atomicMaxF(float* addr, float val) {
    if (val >= 0.f) atomicMax((int*)addr, __float_as_int(val));
    else            atomicMin((unsigned int*)addr, __float_as_uint(val));
}

// ---------------------------------------------------------------- init
// zero accumulators, set segment-max buffers to -FLT_MAX
__global__ void k_init(float* accB, float* outC, float* m1, float* den1,
                       float* m2, float* den2, float* sums, float* cnt) {
    size_t i = (size_t)blockIdx.x * blockDim.x + threadIdx.x;
    if (i < ND) { accB[i] = 0.f; outC[i] = 0.f; }
    if (i < (size_t)NN * HH) { m1[i] = -FLT_MAX; den1[i] = 0.f; }
    if (i < NN) { m2[i] = -FLT_MAX; den2[i] = 0.f; }
    if (i < GG * DD) sums[i] = 0.f;
    if (i < GG) cnt[i] = 0.f;
}

// ---------------------------------------------------------------- GEMM (WMMA f32)
// Y[nrows,128] = X[nrows,128] @ W[128,128]; block = 8 waves, wave w -> col tile w,
// block b -> row tile b. Uses V_WMMA_F32_16X16X4_F32, full f32 precision.
__global__ void k_gemm128(const float* __restrict__ X, const float* __restrict__ W,
                          float* __restrict__ Y) {
    const int lane   = threadIdx.x & 31;
    const int wave   = threadIdx.x >> 5;
    const int rowBase = blockIdx.x * 16;
    const int colBase = wave * 16;
    const int r      = lane & 15;        // row (A) / col (B,D) within tile
    const int khalf  = (lane >> 4) * 2;  // upper 16 lanes handle K+2,K+3

    const float* xrow = X + (size_t)(rowBase + r) * DD;
    v8f acc = {};
#pragma unroll
    for (int k0 = 0; k0 < DD; k0 += 4) {
        v2f a, b;
        a.x = xrow[k0 + khalf];
        a.y = xrow[k0 + khalf + 1];
        b.x = W[(size_t)(k0 + khalf) * DD + colBase + r];
        b.y = W[(size_t)(k0 + khalf + 1) * DD + colBase + r];
        acc = __builtin_amdgcn_wmma_f32_16x16x4_f32(
                  false, a, false, b, (short)0, acc, false, false);
    }
    const int mOff = (lane >> 4) * 8;    // D layout: vgpr i -> M=i (lanes<16) / M=i+8
#pragma unroll
    for (int i = 0; i < 8; ++i)
        Y[(size_t)(rowBase + mOff + i) * DD + colBase + r] = acc[i];
}

// ---------------------------------------------------------------- layer-1 attention
__global__ void k_alpha1(const float* __restrict__ h1,
                         const float* __restrict__ a_src, const float* __restrict__ a_dst,
                         float* __restrict__ asrc, float* __restrict__ adst) {
    int i = blockIdx.x * blockDim.x + threadIdx.x;   // i = n*H + h
    if (i >= NN * HH) return;
    int n = i >> 3, hd = i & 7;
    const float* hp = h1 + (size_t)n * DD + hd * CC;
    const float* as = a_src + hd * CC;
    const float* ad = a_dst + hd * CC;
    float s1 = 0.f, s2 = 0.f;
#pragma unroll
    for (int c = 0; c < CC; ++c) { s1 += hp[c] * as[c]; s2 += hp[c] * ad[c]; }
    asrc[i] = s1; adst[i] = s2;
}

__device__ __forceinline__ void edge_sd(int e, const int* src_e, const int* dst_e,
                                        int& s, int& d) {
    if (e < EE) { s = src_e[e]; d = dst_e[e]; } else { s = d = e - EE; }
}

__global__ void k_edge_max1(const int* __restrict__ src_e, const int* __restrict__ dst_e,
                            const float* __restrict__ asrc, const float* __restrict__ adst,
                            float* __restrict__ m) {
    int e = blockIdx.x * blockDim.x + threadIdx.x;
    if (e >= ETOT) return;
    int s, d; edge_sd(e, src_e, dst_e, s, d);
#pragma unroll
    for (int hd = 0; hd < HH; ++hd)
        atomicMaxF(&m[d * HH + hd], leaky(asrc[s * HH + hd] + adst[d * HH + hd]));
}

__global__ void k_edge_sum1(const int* __restrict__ src_e, const int* __restrict__ dst_e,
                            const float* __restrict__ asrc, const float* __restrict__ adst,
                            const float* __restrict__ m, float* __restrict__ den) {
    int e = blockIdx.x * blockDim.x + threadIdx.x;
    if (e >= ETOT) return;
    int s, d; edge_sd(e, src_e, dst_e, s, d);
#pragma unroll
    for (int hd = 0; hd < HH; ++hd) {
        float v = leaky(asrc[s * HH + hd] + adst[d * HH + hd]);
        atomicAdd(&den[d * HH + hd], expf(v - m[d * HH + hd]));
    }
}

__global__ void k_edge_aggr1(const int* __restrict__ src_e, const int* __restrict__ dst_e,
                             const float* __restrict__ asrc, const float* __restrict__ adst,
                             const float* __restrict__ m, const float* __restrict__ den,
                             const float* __restrict__ h1, float* __restrict__ out) {
    int i = blockIdx.x * blockDim.x + threadIdx.x;   // i = e*H + h
    if (i >= ETOT * HH) return;
    int e = i >> 3, hd = i & 7;
    int s, d; edge_sd(e, src_e, dst_e, s, d);
    float v = leaky(asrc[s * HH + hd] + adst[d * HH + hd]);
    float alpha = expf(v - m[d * HH + hd]) / den[d * HH + hd];
    const float* hs = h1 + (size_t)s * DD + hd * CC;
    float* od = out + (size_t)d * DD + hd * CC;
#pragma unroll
    for (int c = 0; c < CC; ++c) atomicAdd(&od[c], hs[c] * alpha);
}

// ---------------------------------------------------------------- BN(eval) + ELU
__global__ void k_bn_elu(const float* __restrict__ acc, const float* __restrict__ b1,
                         const float* __restrict__ g, const float* __restrict__ b,
                         const float* __restrict__ mu, const float* __restrict__ var,
                         float* __restrict__ outA) {
    size_t i = (size_t)blockIdx.x * blockDim.x + threadIdx.x;
    if (i >= ND) return;
    int dcol = (int)(i & (DD - 1));
    float v = acc[i] + b1[dcol];
    v = (v - mu[dcol]) * g[dcol] * rsqrtf(var[dcol] + BN_EPS) + b[dcol];
    outA[i] = elu(v);
}

// ---------------------------------------------------------------- layer-2 attention
__global__ void k_alpha2(const float* __restrict__ h2,
                         const float* __restrict__ a_src, const float* __restrict__ a_dst,
                         float* __restrict__ asrc, float* __restrict__ adst) {
    int n = blockIdx.x * blockDim.x + threadIdx.x;
    if (n >= NN) return;
    const float* hp = h2 + (size_t)n * DD;
    float s1 = 0.f, s2 = 0.f;
#pragma unroll 4
    for (int c = 0; c < DD; ++c) { s1 += hp[c] * a_src[c]; s2 += hp[c] * a_dst[c]; }
    asrc[n] = s1; adst[n] = s2;
}

__global__ void k_edge_max2(const int* __restrict__ src_e, const int* __restrict__ dst_e,
                            const float* __restrict__ asrc, const float* __restrict__ adst,
                            float* __restrict__ m) {
    int e = blockIdx.x * blockDim.x + threadIdx.x;
    if (e >= ETOT) return;
    int s, d; edge_sd(e, src_e, dst_e, s, d);
    atomicMaxF(&m[d], leaky(asrc[s] + adst[d]));
}

__global__ void k_edge_sum2(const int* __restrict__ src_e, const int* __restrict__ dst_e,
                            const float* __restrict__ asrc, const float* __restrict__ adst,
                            const float* __restrict__ m, float* __restrict__ den) {
    int e = blockIdx.x * blockDim.x + threadIdx.x;
    if (e >= ETOT) return;
    int s, d; edge_sd(e, src_e, dst_e, s, d);
    atomicAdd(&den[d], expf(leaky(asrc[s] + adst[d]) - m[d]));
}

__global__ void k_edge_aggr2(const int* __restrict__ src_e, const int* __restrict__ dst_e,
                             const float* __restrict__ asrc, const float* __restrict__ adst,
                             const float* __restrict__ m, const float* __restrict__ den,
                             const float* __restrict__ h2, float* __restrict__ out) {
    int i = blockIdx.x * blockDim.x + threadIdx.x;   // i = e*8 + group(16 ch)
    if (i >= ETOT * 8) return;
    int e = i >> 3, gp = i & 7;
    int s, d; edge_sd(e, src_e, dst_e, s, d);
    float alpha = expf(leaky(asrc[s] + adst[d]) - m[d]) / den[d];
    const float* hs = h2 + (size_t)s * DD + gp * 16;
    float* od = out + (size_t)d * DD + gp * 16;
#pragma unroll
    for (int c = 0; c < 16; ++c) atomicAdd(&od[c], hs[c] * alpha);
}

// ---------------------------------------------------------------- global mean pool
__global__ void k_pool(const float* __restrict__ out2, const float* __restrict__ b2,
                       const int* __restrict__ batch,
                       float* __restrict__ sums, float* __restrict__ cnt) {
    size_t i = (size_t)blockIdx.x * blockDim.x + threadIdx.x;
    if (i >= ND) return;
    int n = (int)(i >> 7), dcol = (int)(i & (DD - 1));
    int g = batch[n];
    atomicAdd(&sums[(size_t)g * DD + dcol], out2[i] + b2[dcol]);
    if (dcol == 0) atomicAdd(&cnt[g], 1.0f);
}

// ---------------------------------------------------------------- classifier MLP
__global__ void k_mlp1(const float* __restrict__ sums, const float* __restrict__ cnt,
                       const float* __restrict__ Wc1, const float* __restrict__ bc1,
                       float* __restrict__ z) {
    int i = blockIdx.x * blockDim.x + threadIdx.x;   // i = g*64 + j
    if (i >= GG * 64) return;
    int g = i >> 6, j = i & 63;
    float invc = 1.f / fmaxf(cnt[g], 1.f);
    float acc = bc1[j];
#pragma unroll 4
    for (int dcol = 0; dcol < DD; ++dcol)
        acc += (sums[(size_t)g * DD + dcol] * invc) * Wc1[dcol * 64 + j];
    z[i] = elu(acc);
}

__global__ void k_mlp2(const float* __restrict__ z, const float* __restrict__ Wc2,
                       const float* __restrict__ bc2, float* __restrict__ out) {
    int i = blockIdx.x * blockDim.x + threadIdx.x;   // i = g*NC + k
    if (i >= GG * NCLS) return;
    int g = i / NCLS, k = i % NCLS;
    float acc = bc2[k];
#pragma unroll
    for (int j = 0; j < 64; ++j) acc += z[g * 64 + j] * Wc2[j * NCLS + k];
    out[i] = acc;
}

// ---------------------------------------------------------------- launch
static inline int gridFor(size_t n, int blk) { return (int)((n + blk - 1) / blk); }

extern "C" void kernel_launch(void* const* d_in, const int* in_sizes, int n_in,
                              void* d_out, int out_size, void* d_ws, size_t ws_size,
                              hipStream_t stream) {
    const float* x        = (const float*)d_in[0];
    const int*   edge_src = (const int*)  d_in[1];
    const int*   edge_dst = (const int*)  d_in[2];
    const int*   batch    = (const int*)  d_in[3];
    const float* W1       = (const float*)d_in[4];
    const float* b1       = (const float*)d_in[5];
    const float* att_src1 = (const float*)d_in[6];
    const float* att_dst1 = (const float*)d_in[7];
    const float* W2       = (const float*)d_in[8];
    const float* b2       = (const float*)d_in[9];
    const float* att_src2 = (const float*)d_in[10];
    const float* att_dst2 = (const float*)d_in[11];
    const float* bn_g     = (const float*)d_in[12];
    const float* bn_b     = (const float*)d_in[13];
    const float* bn_m     = (const float*)d_in[14];
    const float* bn_v     = (const float*)d_in[15];
    const float* Wc1      = (const float*)d_in[16];
    const float* bc1      = (const float*)d_in[17];
    const float* Wc2      = (const float*)d_in[18];
    const float* bc2      = (const float*)d_in[19];
    float* out = (float*)d_out;

    // workspace layout (floats)
    float* ws = (float*)d_ws;
    float* bufA  = ws;               // h1 -> later h1act (post BN/ELU)
    float* bufB  = ws + ND;          // layer-1 accumulator -> later h2
    float* bufC  = ws + 2 * ND;      // layer-2 accumulator (out2)
    float* asrc1 = ws + 3 * ND;
    float* adst1 = asrc1 + (size_t)NN * HH;
    float* m1    = adst1 + (size_t)NN * HH;
    float* den1  = m1    + (size_t)NN * HH;
    float* asrc2 = den1  + (size_t)NN * HH;
    float* adst2 = asrc2 + NN;
    float* m2    = adst2 + NN;
    float* den2  = m2    + NN;
    float* sums  = den2  + NN;       // G*D
    float* cnt   = sums  + GG * DD;  // G
    float* zbuf  = cnt   + GG;       // G*64

    const int BLK = 256;

    k_init<<<gridFor(ND, BLK), BLK, 0, stream>>>(bufB, bufC, m1, den1, m2, den2, sums, cnt);

    // ---- layer 1
    k_gemm128<<<NN / 16, BLK, 0, stream>>>(x, W1, bufA);                       // h1 = x@W1
    k_alpha1<<<gridFor((size_t)NN * HH, BLK), BLK, 0, stream>>>(bufA, att_src1, att_dst1, asrc1, adst1);
    k_edge_max1<<<gridFor(ETOT, BLK), BLK, 0, stream>>>(edge_src, edge_dst, asrc1, adst1, m1);
    k_edge_sum1<<<gridFor(ETOT, BLK), BLK, 0, stream>>>(edge_src, edge_dst, asrc1, adst1, m1, den1);
    k_edge_aggr1<<<gridFor((size_t)ETOT * HH, BLK), BLK, 0, stream>>>(edge_src, edge_dst,
                                                                      asrc1, adst1, m1, den1, bufA, bufB);
    k_bn_elu<<<gridFor(ND, BLK), BLK, 0, stream>>>(bufB, b1, bn_g, bn_b, bn_m, bn_v, bufA);

    // ---- layer 2
    k_gemm128<<<NN / 16, BLK, 0, stream>>>(bufA, W2, bufB);                    // h2 = h1act@W2
    k_alpha2<<<gridFor(NN, BLK), BLK, 0, stream>>>(bufB, att_src2, att_dst2, asrc2, adst2);
    k_edge_max2<<<gridFor(ETOT, BLK), BLK, 0, stream>>>(edge_src, edge_dst, asrc2, adst2, m2);
    k_edge_sum2<<<gridFor(ETOT, BLK), BLK, 0, stream>>>(edge_src, edge_dst, asrc2, adst2, m2, den2);
    k_edge_aggr2<<<gridFor((size_t)ETOT * 8, BLK), BLK, 0, stream>>>(edge_src, edge_dst,
                                                                     asrc2, adst2, m2, den2, bufB, bufC);

    // ---- pool + classifier
    k_pool<<<gridFor(ND, BLK), BLK, 0, stream>>>(bufC, b2, batch, sums, cnt);
    k_mlp1<<<gridFor(GG * 64, BLK), BLK, 0, stream>>>(sums, cnt, Wc1, bc1, zbuf);
    k_mlp2<<<gridFor(GG * NCLS, BLK), BLK, 0, stream>>>(zbuf, Wc2, bc2, out);
}